// LazyUnite_8632884265500
// MI455X (gfx1250) — compile-verified
//
#include <hip/hip_runtime.h>

typedef __attribute__((ext_vector_type(2))) float v2f;
typedef __attribute__((ext_vector_type(8))) float v8f;

#define C_IN 64
#define C_ROW2 128   // W row length (2*C_IN)

// ---------------------------------------------------------------------------
// Kernel 1: dom_sum[d, :] = sum over edges e with dst[e]==d of x[src[e], :]
// dst is sorted -> one wave per node, binary-search the segment. Deterministic.
// Lane l accumulates channels [2l, 2l+1] as a float2.
// ---------------------------------------------------------------------------
__global__ void seg_sum_kernel(const float* __restrict__ x,
                               const int* __restrict__ src,
                               const int* __restrict__ dst,
                               float* __restrict__ dom,
                               int E, int D) {
    int wave = (int)((blockIdx.x * blockDim.x + threadIdx.x) >> 5);
    int lane = threadIdx.x & 31;
    if (wave >= D) return;
    const int d = wave;

    // lower_bound(dst, d) -- uniform across the wave
    int lo = 0, hi = E;
    while (lo < hi) { int mid = (lo + hi) >> 1; if (dst[mid] < d) lo = mid + 1; else hi = mid; }
    const int start = lo;
    hi = E;
    while (lo < hi) { int mid = (lo + hi) >> 1; if (dst[mid] < d + 1) lo = mid + 1; else hi = mid; }
    const int end = lo;

    const v2f* __restrict__ x2 = (const v2f*)x;
    v2f acc; acc.x = 0.0f; acc.y = 0.0f;
    for (int e = start; e < end; ++e) {
        int s = src[e];
        v2f v = x2[(size_t)s * 32 + lane];
        acc.x += v.x; acc.y += v.y;
    }
    ((v2f*)dom)[(size_t)d * 32 + lane] = acc;
}

// ---------------------------------------------------------------------------
// Kernel 2: node-level GEMMs with fp32 WMMA (16x16x4).
//   grid waves [0, T):    Y1 = dom @ W1^T         (in place over dom)
//   grid waves [T, 2T):   Y2 = x   @ W2^T + b     (into y2)
// One wave computes a 16x64 output tile: 4 N-tiles x 16 K-steps = 64 WMMA.
// A 16x4 f32 layout: lane half h holds cols {k+2h, k+2h+1} of row (lane&15).
// B  4x16 f32 layout: lane half h holds rows {k+2h, k+2h+1} at col (lane&15).
// C/D 16x16 f32: VGPR i -> row i (lanes 0-15) / row i+8 (lanes 16-31).
// ---------------------------------------------------------------------------
__global__ void node_gemm_kernel(const float* __restrict__ x,
                                 float* __restrict__ dom_y1,
                                 float* __restrict__ y2,
                                 const float* __restrict__ Wm,
                                 const float* __restrict__ bias,
                                 int D) {
    int wave = (int)((blockIdx.x * blockDim.x + threadIdx.x) >> 5);
    int lane = threadIdx.x & 31;
    const int T = (D + 15) >> 4;
    if (wave >= 2 * T) return;

    const bool second = (wave >= T);
    const int tile = second ? (wave - T) : wave;
    const float* __restrict__ A = second ? x : dom_y1;
    float* __restrict__ Y = second ? y2 : dom_y1;
    const int woff = second ? C_IN : 0;   // column offset into W rows

    const int m    = lane & 15;
    const int half = lane >> 4;
    int rowA = tile * 16 + m;
    if (rowA >= D) rowA = D - 1;          // safe clamp (D=50000 is 16-aligned anyway)

    // bias (Y2 only), broadcast across the 8 M-rows of each accumulator
    float b0 = 0.0f, b1 = 0.0f, b2 = 0.0f, b3 = 0.0f;
    if (second) {
        b0 = bias[0 * 16 + m]; b1 = bias[1 * 16 + m];
        b2 = bias[2 * 16 + m]; b3 = bias[3 * 16 + m];
    }
    v8f c0, c1, c2, c3;
    for (int i = 0; i < 8; ++i) { c0[i] = b0; c1[i] = b1; c2[i] = b2; c3[i] = b3; }

    const float* __restrict__ Arow = A + (size_t)rowA * C_IN;
    for (int k = 0; k < C_IN; k += 4) {
        const int kk = k + half * 2;
        v2f a = *(const v2f*)(Arow + kk);
        // B[kk..kk+1][n] = W[n][woff + kk .. +1], n = j*16 + m
        v2f bf0 = *(const v2f*)(Wm + (size_t)(0 * 16 + m) * C_ROW2 + woff + kk);
        v2f bf1 = *(const v2f*)(Wm + (size_t)(1 * 16 + m) * C_ROW2 + woff + kk);
        v2f bf2 = *(const v2f*)(Wm + (size_t)(2 * 16 + m) * C_ROW2 + woff + kk);
        v2f bf3 = *(const v2f*)(Wm + (size_t)(3 * 16 + m) * C_ROW2 + woff + kk);
        c0 = __builtin_amdgcn_wmma_f32_16x16x4_f32(false, a, false, bf0, (short)0, c0, false, false);
        c1 = __builtin_amdgcn_wmma_f32_16x16x4_f32(false, a, false, bf1, (short)0, c1, false, false);
        c2 = __builtin_amdgcn_wmma_f32_16x16x4_f32(false, a, false, bf2, (short)0, c2, false, false);
        c3 = __builtin_amdgcn_wmma_f32_16x16x4_f32(false, a, false, bf3, (short)0, c3, false, false);
    }

    // store D-tile: VGPR i -> row tile*16 + i + half*8, col j*16 + m
    for (int i = 0; i < 8; ++i) {
        int row = tile * 16 + i + half * 8;
        if (row < D) {
            float* yr = Y + (size_t)row * C_IN;
            yr[0 * 16 + m] = c0[i];
            yr[1 * 16 + m] = c1[i];
            yr[2 * 16 + m] = c2[i];
            yr[3 * 16 + m] = c3[i];
        }
    }
}

// ---------------------------------------------------------------------------
// Kernel 3: out[e, :] = Y1[dst[e], :] + Y2[src[e], :]
// 16 threads per edge, float4 per thread -> coalesced 256B rows.
// ---------------------------------------------------------------------------
__global__ void edge_combine_kernel(const float* __restrict__ y1,
                                    const float* __restrict__ y2,
                                    const int* __restrict__ src,
                                    const int* __restrict__ dst,
                                    float* __restrict__ out,
                                    long long total) {
    long long t = (long long)blockIdx.x * blockDim.x + threadIdx.x;
    if (t >= total) return;
    int e  = (int)(t >> 4);
    int cg = (int)(t & 15);
    int s = src[e];
    int d = dst[e];
    float4 a = ((const float4*)(y1 + (size_t)d * C_IN))[cg];
    float4 b = ((const float4*)(y2 + (size_t)s * C_IN))[cg];
    float4 o;
    o.x = a.x + b.x; o.y = a.y + b.y; o.z = a.z + b.z; o.w = a.w + b.w;
    ((float4*)out)[t] = o;
}

extern "C" void kernel_launch(void* const* d_in, const int* in_sizes, int n_in,
                              void* d_out, int out_size, void* d_ws, size_t ws_size,
                              hipStream_t stream) {
    const float* x   = (const float*)d_in[0];
    const int*   src = (const int*)d_in[1];
    const int*   dst = (const int*)d_in[2];
    const float* W   = (const float*)d_in[3];
    const float* b   = (const float*)d_in[4];

    const int D = in_sizes[0] / C_IN;   // num nodes == num_domains
    const int E = in_sizes[1];

    float* dom = (float*)d_ws;              // [D,64]  -> becomes Y1 in place
    float* y2  = dom + (size_t)D * C_IN;    // [D,64]

    // 1) segment sum (deterministic, no atomics)
    {
        long long threads = (long long)D * 32;
        int blocks = (int)((threads + 255) / 256);
        seg_sum_kernel<<<blocks, 256, 0, stream>>>(x, src, dst, dom, E, D);
    }
    // 2) node GEMMs via fp32 WMMA (Y1 in place, Y2 + bias)
    {
        int T = (D + 15) >> 4;
        long long threads = (long long)(2 * T) * 32;
        int blocks = (int)((threads + 255) / 256);
        node_gemm_kernel<<<blocks, 256, 0, stream>>>(x, dom, y2, W, b, D);
    }
    // 3) edge combine (HBM-store-bound: 409.6 MB of output)
    {
        long long total = (long long)E * 16;
        int blocks = (int)((total + 255) / 256);
        edge_combine_kernel<<<blocks, 256, 0, stream>>>(dom, y2, src, dst, (float*)d_out, total);
    }
}